// SplineConv_27625229647953
// MI455X (gfx1250) — compile-verified
//
#include <hip/hip_runtime.h>

// Problem constants (match reference)
#define NPTS   4096            // query points  (GEMM M)
#define NDIM   4096            // OUT*IN        (GEMM N)
#define NGRID  64              // 8x8 control grid flattened (GEMM K)

typedef __attribute__((ext_vector_type(2))) float v2f;
typedef __attribute__((ext_vector_type(8))) float v8f;

__device__ __forceinline__ float pick3(float a0, float a1, float a2, int i) {
  // branchless 3-way select (compiles to v_cndmask chain; i clamped by caller's mask)
  return (i == 0) ? a0 : ((i == 1) ? a1 : a2);
}

// ---------------------------------------------------------------------------
// Kernel 1: per point, expand the 3x3 tensor-product B-spline weights into a
// dense 64-float row of W (the GEMM A-matrix).  1 thread per point.
// ---------------------------------------------------------------------------
__global__ void spline_weights_kernel(const float* __restrict__ xy,
                                      const float* __restrict__ Tx,
                                      const float* __restrict__ Ty,
                                      float* __restrict__ Wmat) {
  const int n = blockIdx.x * blockDim.x + threadIdx.x;
  if (n >= NPTS) return;
  const float x = xy[2 * n + 0];
  const float y = xy[2 * n + 1];

  // knot span: kx = searchsorted(T, x, 'right') - 1, clamped to len(T)-p-2 = 7.
  // T[0..2]=lo, T[3..7]=interior (ascending), T[8..10]=hi.
  int kx = 2, ky = 2;
#pragma unroll
  for (int t = 3; t <= 7; ++t) { if (x >= Tx[t]) kx = t; }
#pragma unroll
  for (int t = 3; t <= 7; ++t) { if (y >= Ty[t]) ky = t; }

  // Cox-de Boor basis functions, p = 2 (NURBS-book BasisFuns, unrolled)
  float nx0, nx1, nx2;
  {
    float l1 = x - Tx[kx],     r1 = Tx[kx + 1] - x;
    float l2 = x - Tx[kx - 1], r2 = Tx[kx + 2] - x;
    float t0 = 1.0f / (r1 + l1);
    float a0 = r1 * t0, a1 = l1 * t0;          // j = 1
    float t1 = a0 / (r1 + l2);                 // j = 2, r = 0
    nx0 = r1 * t1;
    float sv = l2 * t1;
    float t2 = a1 / (r2 + l1);                 // j = 2, r = 1
    nx1 = sv + r2 * t2;
    nx2 = l1 * t2;
  }
  float ny0, ny1, ny2;
  {
    float l1 = y - Ty[ky],     r1 = Ty[ky + 1] - y;
    float l2 = y - Ty[ky - 1], r2 = Ty[ky + 2] - y;
    float t0 = 1.0f / (r1 + l1);
    float a0 = r1 * t0, a1 = l1 * t0;
    float t1 = a0 / (r1 + l2);
    ny0 = r1 * t1;
    float sv = l2 * t1;
    float t2 = a1 / (r2 + l1);
    ny1 = sv + r2 * t2;
    ny2 = l1 * t2;
  }

  const int bx = kx - 2, by = ky - 2;
  float4* out4 = reinterpret_cast<float4*>(Wmat + (size_t)n * NGRID);
#pragma unroll
  for (int q = 0; q < 16; ++q) {
    float v[4];
#pragma unroll
    for (int e = 0; e < 4; ++e) {
      const int idx = q * 4 + e;              // flat grid index = row*8 + col
      const int i = (idx >> 3) - bx;          // x-direction (H axis)
      const int j = (idx & 7) - by;           // y-direction (W axis)
      const bool ok = ((unsigned)i < 3u) && ((unsigned)j < 3u);
      const float wv = pick3(nx0, nx1, nx2, i) * pick3(ny0, ny1, ny2, j);
      v[e] = ok ? wv : 0.0f;
    }
    out4[q] = make_float4(v[0], v[1], v[2], v[3]);
  }
}

// ---------------------------------------------------------------------------
// Kernel 2: Out[M=4096, N=4096] = W[M, 64] x C^T[64, N] via f32 WMMA.
// One wave32 per 32x32 output block = 2x2 register tiles of 16x16, K = 64.
// Per K-chunk: 4 x b64 loads feed 4 x V_WMMA_F32_16X16X4_F32 (load:wmma = 1),
// and each A/B fragment is reused across two tiles (halves L2 refetch).
//
// Operand layouts (CDNA5 ISA 7.12.2):
//   A 16x4 f32 : lanes 0-15 -> M=lane,  VGPR0=K(kk+0), VGPR1=K(kk+1)
//                lanes16-31 -> M=lane-16, VGPR0=K(kk+2), VGPR1=K(kk+3)
//   B 4x16 f32 : lanes 0-15 -> N=lane,  VGPR0=K(kk+0), VGPR1=K(kk+1)
//                lanes16-31 -> N=lane-16, VGPR0=K(kk+2), VGPR1=K(kk+3)
//   C/D 16x16  : VGPR v -> M=v (lanes 0-15) / M=v+8 (lanes 16-31), N=lane%16
// ---------------------------------------------------------------------------
__global__ void spline_gemm_kernel(const float* __restrict__ Wmat,  // [NPTS][64]
                                   const float* __restrict__ C,     // [NDIM][64]
                                   float* __restrict__ out) {       // [NPTS][NDIM]
  const int wave = (int)((blockIdx.x * blockDim.x + threadIdx.x) >> 5);
  const int lane = (int)(threadIdx.x & 31);
  const int tileN = wave & (NDIM / 32 - 1);   // along channel dim (128 blocks)
  const int tileM = wave >> 7;                // along points      (128 blocks)
  const int row0 = tileM * 32;
  const int col0 = tileN * 32;
  const int half = lane >> 4;                 // 0: K lo pair, 1: K hi pair
  const int l16  = lane & 15;

  const float* Arow0 = Wmat + (size_t)(row0 + l16) * NGRID + 2 * half;
  const float* Arow1 = Arow0 + (size_t)16 * NGRID;
  const float* Brow0 = C + (size_t)(col0 + l16) * NGRID + 2 * half;  // B[k][n] = C[n*64+k]
  const float* Brow1 = Brow0 + (size_t)16 * NGRID;

  v8f acc00 = {}, acc01 = {}, acc10 = {}, acc11 = {};
#pragma unroll
  for (int c = 0; c < 16; ++c) {
    v2f a0 = *reinterpret_cast<const v2f*>(Arow0 + 4 * c);
    v2f a1 = *reinterpret_cast<const v2f*>(Arow1 + 4 * c);
    v2f b0 = *reinterpret_cast<const v2f*>(Brow0 + 4 * c);
    v2f b1 = *reinterpret_cast<const v2f*>(Brow1 + 4 * c);
    // 8 args: (neg_a, A, neg_b, B, c_mod, C, reuse_a, reuse_b)
    acc00 = __builtin_amdgcn_wmma_f32_16x16x4_f32(false, a0, false, b0, (short)0,
                                                  acc00, false, false);
    acc01 = __builtin_amdgcn_wmma_f32_16x16x4_f32(false, a0, false, b1, (short)0,
                                                  acc01, false, false);
    acc10 = __builtin_amdgcn_wmma_f32_16x16x4_f32(false, a1, false, b0, (short)0,
                                                  acc10, false, false);
    acc11 = __builtin_amdgcn_wmma_f32_16x16x4_f32(false, a1, false, b1, (short)0,
                                                  acc11, false, false);
  }

  // Store the 2x2 tiles per the C/D layout.
  float* o00 = out + (size_t)row0 * NDIM + (size_t)col0 + l16;
  float* o01 = o00 + 16;
  float* o10 = o00 + (size_t)16 * NDIM;
  float* o11 = o10 + 16;
#pragma unroll
  for (int v = 0; v < 8; ++v) {
    const size_t roff = (size_t)(v + 8 * half) * NDIM;
    o00[roff] = acc00[v];
    o01[roff] = acc01[v];
    o10[roff] = acc10[v];
    o11[roff] = acc11[v];
  }
}

// ---------------------------------------------------------------------------
// Host launcher.  Inputs (setup_inputs order): xy[4096,2] f32, C[64,64,8,8] f32,
// Tx[11] f32, Ty[11] f32.  Output: f32 [4096, 64, 64] flat = [4096, 4096].
// d_ws holds W[4096][64] f32 = 1 MB.
// ---------------------------------------------------------------------------
extern "C" void kernel_launch(void* const* d_in, const int* in_sizes, int n_in,
                              void* d_out, int out_size, void* d_ws, size_t ws_size,
                              hipStream_t stream) {
  const float* xy = (const float*)d_in[0];
  const float* C  = (const float*)d_in[1];
  const float* Tx = (const float*)d_in[2];
  const float* Ty = (const float*)d_in[3];
  float* out  = (float*)d_out;
  float* Wmat = (float*)d_ws;   // NPTS * NGRID * 4 = 1 MB

  spline_weights_kernel<<<NPTS / 256, 256, 0, stream>>>(xy, Tx, Ty, Wmat);

  const int total_waves = (NPTS / 32) * (NDIM / 32);   // 16384 blocks of 32x32
  const int threads = 256;                             // 8 waves per block
  const int blocks = total_waves * 32 / threads;       // 2048
  spline_gemm_kernel<<<blocks, threads, 0, stream>>>(Wmat, C, out);
}